// SoapBubble_76802605187323
// MI455X (gfx1250) — compile-verified
//
#include <hip/hip_runtime.h>
#include <math.h>

// ---------------- problem constants (from reference) ----------------
#define N_CELLS 128
#define V_PER   64
#define F_PER   124
#define N_V     (N_CELLS * V_PER)     // 8192
#define N_F     (N_CELLS * F_PER)     // 15872
#define N_ITERS 5
#define UF_ITERS 8
#define TIME_STEPF 1e-05f
#define THRESHF 0.3f
#define BULKF 2500.0f
#define HALF_T2 (0.5f * THRESHF * THRESHF)

#define N_TILES    512                // 8192 / 16
#define TILES_J_PER_WAVE 16
#define ROW_WORDS  256                // 8192 bits / 32
#define ROW_SHORTS 512                // 8192 bits / 16

typedef __attribute__((ext_vector_type(2))) float v2f;
typedef __attribute__((ext_vector_type(8))) float v8f;

// ---------------- small float3 helpers ----------------
struct F3 { float x, y, z; };
__device__ __forceinline__ F3 operator+(F3 a, F3 b){ return {a.x+b.x, a.y+b.y, a.z+b.z}; }
__device__ __forceinline__ F3 operator-(F3 a, F3 b){ return {a.x-b.x, a.y-b.y, a.z-b.z}; }
__device__ __forceinline__ F3 operator*(F3 a, float s){ return {a.x*s, a.y*s, a.z*s}; }
__device__ __forceinline__ float dot3(F3 a, F3 b){ return a.x*b.x + a.y*b.y + a.z*b.z; }
__device__ __forceinline__ F3 cross3(F3 a, F3 b){
  return {a.y*b.z - a.z*b.y, a.z*b.x - a.x*b.z, a.x*b.y - a.y*b.x};
}
__device__ __forceinline__ F3 ld3(const float* p, int i){ return {p[3*i], p[3*i+1], p[3*i+2]}; }

// ---------------- scalar parameter prep ----------------
__global__ void k_params(const float* np, const float* nst, float* params) {
  if (threadIdx.x == 0 && blockIdx.x == 0) {
    float pressure = np[0] * (200.0f - 10.0f) + 10.0f;
    float st       = nst[0] * (0.02f - 0.002f) + 0.002f;
    params[0] = st;
    params[1] = expf(pressure / BULKF);   // target_vol
  }
}

// ---------------- cell-bbox contact table ----------------
__global__ void k_cellc(const float* bb, int* cellc) {
  int idx = blockIdx.x * blockDim.x + threadIdx.x;
  if (idx >= N_CELLS * N_CELLS) return;
  int i = idx / N_CELLS, j = idx % N_CELLS;
  const float* bi = bb + i * 6;
  const float* bj = bb + j * 6;
  bool c = true;
  #pragma unroll
  for (int k = 0; k < 3; ++k)
    c = c && (bi[k] <= bj[3 + k]) && (bj[k] <= bi[3 + k]);
  cellc[idx] = (c && (i != j)) ? 1 : 0;
}

// ---------------- pack verts as (x,y,z, 0.5*|v|^2) for WMMA ----------------
__global__ void k_pack(const float* vcur, float4* v4) {
  int i = blockIdx.x * blockDim.x + threadIdx.x;
  if (i >= N_V) return;
  float x = vcur[3*i + 0], y = vcur[3*i + 1], z = vcur[3*i + 2];
  v4[i] = make_float4(x, y, z, 0.5f * (x*x + y*y + z*z));
}

// ---------------- contact matrix via V_WMMA_F32_16X16X4_F32 ----------------
// One wave computes a 16x16 tile.  The row bias is folded into the matmul:
//   A row m = (x_m, y_m, z_m, -a_m),  B col n = (x_n, y_n, z_n, 1)
//   D[m][n] = v_m . v_n - a_m        (a = 0.5*|v|^2)
// Contact(m,n) <=> d2 < T^2 <=> D[m][n] > a_n - T^2/2, and a_n is this lane's
// own vb.w, so the compare needs no cross-lane traffic at all.
//
// 32-bit A 16x4 layout: lanes 0-15 hold (M=lane, K=0/K=1) in VGPR0/1,
// lanes 16-31 hold (M=lane-16, K=2/K=3).  B 4x16 mirrors it.
__global__ void k_contact(const float4* __restrict__ verts4,
                          const int* __restrict__ cellc,
                          unsigned short* __restrict__ contact16) {
  int wave = (blockIdx.x * blockDim.x + threadIdx.x) >> 5;
  int lane = threadIdx.x & 31;
  int ti = wave / (N_TILES / TILES_J_PER_WAVE);
  int tjBase = (wave % (N_TILES / TILES_J_PER_WAVE)) * TILES_J_PER_WAVE;
  if (ti >= N_TILES) return;

  const int i0 = ti * 16;
  const int ci = i0 >> 6;                 // 64 verts per cell, tiles 16-aligned
  float4 va = verts4[i0 + (lane & 15)];
  v2f A;
  A.x = (lane < 16) ? va.x : va.z;        // K0 / K2
  A.y = (lane < 16) ? va.y : -va.w;       // K1 / K3 = -a_m  (bias fold)

  for (int tjo = 0; tjo < TILES_J_PER_WAVE; ++tjo) {
    int tj = tjBase + tjo;
    int j0 = tj * 16;
    int cj = j0 >> 6;
    if (cellc[ci * N_CELLS + cj] == 0) {  // wave-uniform: tile fully disabled
      if (lane < 16)
        contact16[(size_t)(i0 + lane) * ROW_SHORTS + tj] = 0;
      continue;
    }
    float4 vb = verts4[j0 + (lane & 15)];
    v2f B;
    B.x = (lane < 16) ? vb.x : vb.z;      // K0 / K2
    B.y = (lane < 16) ? vb.y : 1.0f;      // K1 / K3 = 1 (bias fold)

    v8f acc = {};
    // D[m][n] = dot(v_{i0+m}, v_{j0+n}) - a_{i0+m}, full f32
    acc = __builtin_amdgcn_wmma_f32_16x16x4_f32(
        /*neg_a=*/false, A, /*neg_b=*/false, B,
        /*c_mod=*/(short)0, acc, /*reuse_a=*/false, /*reuse_b=*/false);

    float thr = vb.w - HALF_T2;           // a_n - T^2/2 (lane-local)
    unsigned int bb[8];
    #pragma unroll
    for (int r = 0; r < 8; ++r) {
      // C/D layout: lane<16 -> M=r, lane>=16 -> M=r+8 ; N = lane&15
      bool pred = acc[r] > thr;           // <=> d2 < T^2
      bb[r] = __builtin_amdgcn_ballot_w32(pred);
      // bits[15:0] -> row i0+r, cols j0+0..15 ; bits[31:16] -> row i0+r+8
    }
    if (lane < 16) {
      unsigned int sel = bb[0];
      #pragma unroll
      for (int r = 1; r < 8; ++r)
        if ((lane & 7) == r) sel = bb[r];
      unsigned short o = (lane < 8) ? (unsigned short)(sel & 0xFFFFu)
                                    : (unsigned short)(sel >> 16);
      contact16[(size_t)(i0 + lane) * ROW_SHORTS + tj] = o;
    }
  }
}

// ---------------- union-find ----------------
__global__ void k_init_labels(int* labels) {
  int i = blockIdx.x * blockDim.x + threadIdx.x;
  if (i < N_V) labels[i] = i;
}

// one block (256 threads) per row i: nm = min over set bits of labels[j]
__global__ void k_uf_min(const unsigned int* __restrict__ contactW,
                         const int* __restrict__ labels,
                         int* __restrict__ tmpl) {
  __shared__ int red[256];
  int i = blockIdx.x;
  int t = threadIdx.x;
  unsigned int w = contactW[(size_t)i * ROW_WORDS + t];
  int m = N_V;
  while (w) {
    int b = __ffs(w) - 1;
    w &= w - 1u;
    int lj = labels[t * 32 + b];
    m = (lj < m) ? lj : m;
  }
  red[t] = m;
  __syncthreads();
  for (int s = 128; s > 0; s >>= 1) {
    if (t < s) red[t] = min(red[t], red[t + s]);
    __syncthreads();
  }
  if (t == 0) tmpl[i] = min(labels[i], red[0]);
}

// labels = tmpl[tmpl[]]  (pointer jumping)
__global__ void k_uf_jump(const int* __restrict__ tmpl, int* __restrict__ labels) {
  int i = blockIdx.x * blockDim.x + threadIdx.x;
  if (i < N_V) labels[i] = tmpl[tmpl[i]];
}

// ---------------- segment mean machinery ----------------
__global__ void k_zero(float* p, int n) {
  int i = blockIdx.x * blockDim.x + threadIdx.x;
  if (i < n) p[i] = 0.0f;
}

__global__ void k_scatter3(const float* __restrict__ src,
                           const int* __restrict__ labels,
                           float* __restrict__ sums) {
  int i = blockIdx.x * blockDim.x + threadIdx.x;
  if (i >= N_V) return;
  int l = labels[i];
  atomicAdd(&sums[4*l + 0], src[3*i + 0]);
  atomicAdd(&sums[4*l + 1], src[3*i + 1]);
  atomicAdd(&sums[4*l + 2], src[3*i + 2]);
  atomicAdd(&sums[4*l + 3], 1.0f);
}

__global__ void k_gather_mean(const float* __restrict__ sums,
                              const int* __restrict__ labels,
                              float* __restrict__ dst) {
  int i = blockIdx.x * blockDim.x + threadIdx.x;
  if (i >= N_V) return;
  int l = labels[i];
  float c = fmaxf(sums[4*l + 3], 1.0f);
  dst[3*i + 0] = sums[4*l + 0] / c;
  dst[3*i + 1] = sums[4*l + 1] / c;
  dst[3*i + 2] = sums[4*l + 2] / c;
}

// vcur += dt * group_mean(forces)
__global__ void k_update(const float* __restrict__ sums,
                         const int* __restrict__ labels,
                         float* __restrict__ vcur) {
  int i = blockIdx.x * blockDim.x + threadIdx.x;
  if (i >= N_V) return;
  int l = labels[i];
  float c = fmaxf(sums[4*l + 3], 1.0f);
  vcur[3*i + 0] += TIME_STEPF * sums[4*l + 0] / c;
  vcur[3*i + 1] += TIME_STEPF * sums[4*l + 1] / c;
  vcur[3*i + 2] += TIME_STEPF * sums[4*l + 2] / c;
}

// ---------------- energy gradient ----------------
__global__ void k_centroid(const float* __restrict__ vcur, float* __restrict__ cen) {
  __shared__ float sx[64], sy[64], sz[64];
  int c = blockIdx.x, t = threadIdx.x;
  F3 v = ld3(vcur, c * V_PER + t);
  sx[t] = v.x; sy[t] = v.y; sz[t] = v.z;
  __syncthreads();
  for (int s = 32; s > 0; s >>= 1) {
    if (t < s) { sx[t] += sx[t+s]; sy[t] += sy[t+s]; sz[t] += sz[t+s]; }
    __syncthreads();
  }
  if (t == 0) {
    cen[3*c + 0] = sx[0] * (1.0f / V_PER);
    cen[3*c + 1] = sy[0] * (1.0f / V_PER);
    cen[3*c + 2] = sz[0] * (1.0f / V_PER);
  }
}

// per-cell: vol, coef = BULK*(vol - target), and centroid chain-rule force
// dvol/dc = -(1/6) * sum_f (p1xp2 + p2xp0 + p0xp1);  per-vertex force from
// centroid term = -coef*dvol/dc / V_PER = coef * S / (6*V_PER)
__global__ void k_cellvol(const float* __restrict__ vcur,
                          const int* __restrict__ cfl,
                          const int* __restrict__ faces,
                          const float* __restrict__ cen,
                          const float* __restrict__ params,
                          float* __restrict__ coef,
                          float* __restrict__ fcent) {
  __shared__ float r0[128], r1[128], r2[128], r3[128];
  int c = blockIdx.x, t = threadIdx.x;
  F3 cc = ld3(cen, c);
  float tp = 0.0f; F3 s = {0.f, 0.f, 0.f};
  if (t < F_PER) {
    int f = cfl[c * F_PER + t];
    F3 p0 = ld3(vcur, faces[3*f + 0]) - cc;
    F3 p1 = ld3(vcur, faces[3*f + 1]) - cc;
    F3 p2 = ld3(vcur, faces[3*f + 2]) - cc;
    F3 c12 = cross3(p1, p2);
    tp = dot3(p0, c12);
    s = c12 + cross3(p2, p0) + cross3(p0, p1);
  }
  r0[t] = tp; r1[t] = s.x; r2[t] = s.y; r3[t] = s.z;
  __syncthreads();
  for (int k = 64; k > 0; k >>= 1) {
    if (t < k) { r0[t]+=r0[t+k]; r1[t]+=r1[t+k]; r2[t]+=r2[t+k]; r3[t]+=r3[t+k]; }
    __syncthreads();
  }
  if (t == 0) {
    float vol = r0[0] * (1.0f / 6.0f);
    float co = BULKF * (vol - params[1]);
    coef[c] = co;
    float sc = co / (6.0f * (float)V_PER);
    fcent[3*c + 0] = r1[0] * sc;
    fcent[3*c + 1] = r2[0] * sc;
    fcent[3*c + 2] = r3[0] * sc;
  }
}

// per-face: surface-tension area gradient + direct volume gradient, atomics
__global__ void k_facegrad(const float* __restrict__ vcur,
                           const int* __restrict__ faces,
                           const int* __restrict__ faceCell,
                           const float* __restrict__ cen,
                           const float* __restrict__ coef,
                           const float* __restrict__ params,
                           float* __restrict__ F) {
  int f = blockIdx.x * blockDim.x + threadIdx.x;
  if (f >= N_F) return;
  float st = params[0];
  int i0 = faces[3*f + 0], i1 = faces[3*f + 1], i2 = faces[3*f + 2];
  F3 v0 = ld3(vcur, i0), v1 = ld3(vcur, i1), v2 = ld3(vcur, i2);
  F3 n = cross3(v1 - v0, v2 - v0);
  float nn = dot3(n, n) + 1e-12f;
  float sA = 0.5f * st * rsqrtf(nn);      // d(st*A)/dn chain factor
  F3 g0 = cross3(v1 - v2, n) * sA;
  F3 g1 = cross3(v2 - v0, n) * sA;
  F3 g2 = cross3(v0 - v1, n) * sA;
  int c = faceCell[f];
  float co = coef[c] * (1.0f / 6.0f);
  F3 cc = ld3(cen, c);
  F3 p0 = v0 - cc, p1 = v1 - cc, p2 = v2 - cc;
  g0 = g0 + cross3(p1, p2) * co;
  g1 = g1 + cross3(p2, p0) * co;
  g2 = g2 + cross3(p0, p1) * co;
  atomicAdd(&F[3*i0+0], -g0.x); atomicAdd(&F[3*i0+1], -g0.y); atomicAdd(&F[3*i0+2], -g0.z);
  atomicAdd(&F[3*i1+0], -g1.x); atomicAdd(&F[3*i1+1], -g1.y); atomicAdd(&F[3*i1+2], -g1.z);
  atomicAdd(&F[3*i2+0], -g2.x); atomicAdd(&F[3*i2+1], -g2.y); atomicAdd(&F[3*i2+2], -g2.z);
}

__global__ void k_spread(const float* __restrict__ fcent,
                         const int* __restrict__ vertexCell,
                         float* __restrict__ F) {
  int i = blockIdx.x * blockDim.x + threadIdx.x;
  if (i >= N_V) return;
  int c = vertexCell[i];
  F[3*i + 0] += fcent[3*c + 0];
  F[3*i + 1] += fcent[3*c + 1];
  F[3*i + 2] += fcent[3*c + 2];
}

// ---------------- host orchestration ----------------
extern "C" void kernel_launch(void* const* d_in, const int* in_sizes, int n_in,
                              void* d_out, int out_size, void* d_ws, size_t ws_size,
                              hipStream_t stream) {
  (void)in_sizes; (void)n_in; (void)out_size; (void)ws_size;
  const float* vertices     = (const float*)d_in[0];
  const float* bboxes       = (const float*)d_in[1];
  const float* np           = (const float*)d_in[2];
  const float* nst          = (const float*)d_in[3];
  const int*   faces        = (const int*)d_in[4];
  const int*   faceCell     = (const int*)d_in[5];
  const int*   cellFaceList = (const int*)d_in[6];
  const int*   vertexCell   = (const int*)d_in[7];
  float* out = (float*)d_out;

  // workspace carving (256B aligned)
  char* w = (char*)d_ws;
  auto carve = [&](size_t bytes) -> char* {
    char* p = w;
    w += (bytes + 255) & ~(size_t)255;
    return p;
  };
  float4* verts4 = (float4*)carve((size_t)N_V * sizeof(float4));
  float*  vcur   = (float*) carve((size_t)N_V * 3 * sizeof(float));
  float*  fbuf   = (float*) carve((size_t)N_V * 3 * sizeof(float));
  float*  sums   = (float*) carve((size_t)N_V * 4 * sizeof(float));
  int*    labels = (int*)   carve((size_t)N_V * sizeof(int));
  int*    tmpl   = (int*)   carve((size_t)N_V * sizeof(int));
  int*    cellc  = (int*)   carve((size_t)N_CELLS * N_CELLS * sizeof(int));
  float*  cen    = (float*) carve((size_t)N_CELLS * 3 * sizeof(float));
  float*  coef   = (float*) carve((size_t)N_CELLS * sizeof(float));
  float*  fcent  = (float*) carve((size_t)N_CELLS * 3 * sizeof(float));
  float*  params = (float*) carve(16 * sizeof(float));
  unsigned int* contact = (unsigned int*)carve((size_t)N_V * ROW_WORDS * sizeof(unsigned int));

  hipMemcpyAsync(vcur, vertices, (size_t)N_V * 3 * sizeof(float),
                 hipMemcpyDeviceToDevice, stream);
  k_params<<<1, 32, 0, stream>>>(np, nst, params);
  k_cellc<<<(N_CELLS * N_CELLS + 255) / 256, 256, 0, stream>>>(bboxes, cellc);

  const int B = 256;
  const int gV   = (N_V + B - 1) / B;           // 32
  const int gV3  = (N_V * 3 + B - 1) / B;       // 96
  const int gV4  = (N_V * 4 + B - 1) / B;       // 128
  const int gF   = (N_F + B - 1) / B;
  // contact: 512 tile-rows * 32 wave-jobs/row = 16384 waves, 8 waves/block
  const int gCt  = (N_TILES * (N_TILES / TILES_J_PER_WAVE)) / 8;  // 2048

  for (int it = 0; it < N_ITERS; ++it) {
    // 1) contact matrix via WMMA
    k_pack<<<gV, B, 0, stream>>>(vcur, verts4);
    k_contact<<<gCt, B, 0, stream>>>(verts4, cellc, (unsigned short*)contact);

    // 2) union-find
    k_init_labels<<<gV, B, 0, stream>>>(labels);
    for (int u = 0; u < UF_ITERS; ++u) {
      k_uf_min<<<N_V, B, 0, stream>>>(contact, labels, tmpl);
      k_uf_jump<<<gV, B, 0, stream>>>(tmpl, labels);
    }

    // 3) verts = group_mean(verts, labels)
    k_zero<<<gV4, B, 0, stream>>>(sums, N_V * 4);
    k_scatter3<<<gV, B, 0, stream>>>(vcur, labels, sums);
    k_gather_mean<<<gV, B, 0, stream>>>(sums, labels, vcur);

    // 4) forces = -grad(E)
    k_zero<<<gV3, B, 0, stream>>>(fbuf, N_V * 3);
    k_centroid<<<N_CELLS, V_PER, 0, stream>>>(vcur, cen);
    k_cellvol<<<N_CELLS, 128, 0, stream>>>(vcur, cellFaceList, faces, cen,
                                           params, coef, fcent);
    k_facegrad<<<gF, B, 0, stream>>>(vcur, faces, faceCell, cen, coef, params, fbuf);
    k_spread<<<gV, B, 0, stream>>>(fcent, vertexCell, fbuf);

    // 5) verts += dt * group_mean(forces, labels)
    k_zero<<<gV4, B, 0, stream>>>(sums, N_V * 4);
    k_scatter3<<<gV, B, 0, stream>>>(fbuf, labels, sums);
    k_update<<<gV, B, 0, stream>>>(sums, labels, vcur);
  }

  hipMemcpyAsync(out, vcur, (size_t)N_V * 3 * sizeof(float),
                 hipMemcpyDeviceToDevice, stream);
}